// ReadOutModel_40432822124776
// MI455X (gfx1250) — compile-verified
//
#include <hip/hip_runtime.h>

// Linear RNN: u = x@W_ih^T + b_ih ; h_t = h_{t-1}@W_hh^T + u_{t-1} ; out = h@W_out^T + b_out
// fp32 end-to-end using V_WMMA_F32_16X16X4_F32 (16x16 tiles, K=4 per op).

typedef float v2f __attribute__((ext_vector_type(2)));
typedef float v8f __attribute__((ext_vector_type(8)));

#define RNN_N 32
#define RNN_T 4096
#define RNN_I 88
#define RNN_H 256
#define RNN_O 88

__device__ __forceinline__ v8f wmma_f32_k4(v2f a, v2f b, v8f c) {
  // 8 args: (neg_a, A, neg_b, B, c_mod, C, reuse_a, reuse_b)
  return __builtin_amdgcn_wmma_f32_16x16x4_f32(false, a, false, b, (short)0, c,
                                               false, false);
}

// ---------------------------------------------------------------------------
// Kernel A: u[r, h] = x[r, :] @ W_ih^T + b_ih   (r = n*T + t, 131072 rows)
// Writes u into the hiddens region of d_out (in-place converted by k_scan).
// ---------------------------------------------------------------------------
__global__ __launch_bounds__(512) void k_proj_u(const float* __restrict__ x,
                                                const float* __restrict__ W_ih,
                                                const float* __restrict__ b_ih,
                                                float* __restrict__ u) {
  const int wave = threadIdx.x >> 5;       // 0..15 : nj
  const int lane = threadIdx.x & 31;
  const int hi   = lane >> 4;              // half-wave
  const int ln   = lane & 15;
  const long mt  = blockIdx.x;             // 0..8191 row tiles
  const int col  = wave * 16 + ln;         // hidden col 0..255

  const float* xrow = x   + (mt * 16 + ln) * RNN_I + 2 * hi;  // A: row=ln, K=2*hi+{0,1}
  const float* wrow = W_ih + (long)col * RNN_I + 2 * hi;      // B: col=ln, K=2*hi+{0,1}

  const float bias = b_ih[col];
  v8f c;
#pragma unroll
  for (int v = 0; v < 8; ++v) c[v] = bias;

#pragma unroll
  for (int ks = 0; ks < RNN_I / 4; ++ks) {                    // 22 k-steps (88 = 22*4)
    v2f a = *(const v2f*)(xrow + ks * 4);
    v2f b = *(const v2f*)(wrow + ks * 4);
    c = wmma_f32_k4(a, b, c);
  }

  float* urow = u + (mt * 16 + 8 * hi) * RNN_H + col;         // D: row = 8*hi + v
#pragma unroll
  for (int v = 0; v < 8; ++v) urow[(long)v * RNN_H] = c[v];
}

// ---------------------------------------------------------------------------
// Kernel B: sequential scan, single workgroup, 16 waves (512 threads).
// Wave nj owns BOTH 16-row tiles (mi=0,1) of columns nj*16..nj*16+15, so its
// 64 register-resident W_hh B-fragments (16 cols x 256 K) serve two output
// tiles. ~212 live VGPRs/wave at 4 waves/SIMD -> fits without spilling.
// h double-buffered in LDS (64 KB). hid region holds u on entry, h on exit.
// ---------------------------------------------------------------------------
__global__ __launch_bounds__(512) void k_scan(const float* __restrict__ h0,
                                              const float* __restrict__ W_hh,
                                              float* __restrict__ hid) {
  __shared__ float hbuf[2][RNN_N][RNN_H];                     // 64 KB

  const int nj   = threadIdx.x >> 5;   // 0..15 (hidden col tile)
  const int lane = threadIdx.x & 31;
  const int hi   = lane >> 4;
  const int ln   = lane & 15;
  const int col  = nj * 16 + ln;

  // Loop-invariant B fragments: bf[ks] = W_hh[col][4*ks + 2*hi + {0,1}]
  v2f bf[RNN_H / 4];
  {
    const float* wrow = W_hh + (long)col * RNN_H + 2 * hi;
#pragma unroll
    for (int ks = 0; ks < RNN_H / 4; ++ks) bf[ks] = *(const v2f*)(wrow + ks * 4);
  }

  const int nb0 = 8 * hi;              // mi=0 tile: rows nb0 + v   (v = 0..7)
  const int nb1 = 16 + 8 * hi;         // mi=1 tile: rows nb1 + v

  // t = 0: h_first = h0 + u[:,0]; keep u[:,0] in registers for step t=1.
  float ucurA[8], ucurB[8];
#pragma unroll
  for (int v = 0; v < 8; ++v) {
    {
      const int n = nb0 + v;
      const float u0 = hid[((long)n * RNN_T + 0) * RNN_H + col];
      ucurA[v] = u0;
      const float hf = h0[n * RNN_H + col] + u0;
      hbuf[0][n][col] = hf;
      hid[((long)n * RNN_T + 0) * RNN_H + col] = hf;
    }
    {
      const int n = nb1 + v;
      const float u0 = hid[((long)n * RNN_T + 0) * RNN_H + col];
      ucurB[v] = u0;
      const float hf = h0[n * RNN_H + col] + u0;
      hbuf[0][n][col] = hf;
      hid[((long)n * RNN_T + 0) * RNN_H + col] = hf;
    }
  }
  __syncthreads();

  int cur = 0;
  for (int t = 1; t < RNN_T; ++t) {
    // Prefetch u[:,t] (slot t still holds u; we overwrite it below).
    float unextA[8], unextB[8];
#pragma unroll
    for (int v = 0; v < 8; ++v) {
      unextA[v] = hid[((long)(nb0 + v) * RNN_T + t) * RNN_H + col];
      unextB[v] = hid[((long)(nb1 + v) * RNN_T + t) * RNN_H + col];
    }

    v8f cA0, cA1, cB0, cB1;
#pragma unroll
    for (int v = 0; v < 8; ++v) {
      cA0[v] = ucurA[v]; cA1[v] = 0.f;
      cB0[v] = ucurB[v]; cB1[v] = 0.f;
    }

    const float* hrow0 = &hbuf[cur][0][0] + (long)ln * RNN_H + 2 * hi;        // mi=0
    const float* hrow1 = &hbuf[cur][0][0] + (long)(16 + ln) * RNN_H + 2 * hi; // mi=1
#pragma unroll
    for (int ks = 0; ks < RNN_H / 4; ks += 2) {               // 128 WMMAs, 4 chains
      v2f a00 = *(const v2f*)(hrow0 + (ks + 0) * 4);
      v2f a10 = *(const v2f*)(hrow1 + (ks + 0) * 4);
      v2f a01 = *(const v2f*)(hrow0 + (ks + 1) * 4);
      v2f a11 = *(const v2f*)(hrow1 + (ks + 1) * 4);
      cA0 = wmma_f32_k4(a00, bf[ks + 0], cA0);
      cB0 = wmma_f32_k4(a10, bf[ks + 0], cB0);
      cA1 = wmma_f32_k4(a01, bf[ks + 1], cA1);
      cB1 = wmma_f32_k4(a11, bf[ks + 1], cB1);
    }

    // u[:,t] loads must land before h_t overwrites slot t (load->store, same addr).
    asm volatile("s_wait_loadcnt 0x0" ::: "memory");

#pragma unroll
    for (int v = 0; v < 8; ++v) {
      {
        const float hval = cA0[v] + cA1[v];
        const int n = nb0 + v;
        hbuf[cur ^ 1][n][col] = hval;
        hid[((long)n * RNN_T + t) * RNN_H + col] = hval;
        ucurA[v] = unextA[v];
      }
      {
        const float hval = cB0[v] + cB1[v];
        const int n = nb1 + v;
        hbuf[cur ^ 1][n][col] = hval;
        hid[((long)n * RNN_T + t) * RNN_H + col] = hval;
        ucurB[v] = unextB[v];
      }
    }
    cur ^= 1;
    __syncthreads();
  }
}

// ---------------------------------------------------------------------------
// Kernel C: out[r, o] = hiddens[r, :] @ W_out^T + b_out. O=88 -> 6 col tiles
// (last tile half-masked, masking kept outside the WMMA so EXEC is all-ones).
// ---------------------------------------------------------------------------
__global__ __launch_bounds__(192) void k_readout(const float* __restrict__ hid,
                                                 const float* __restrict__ W_out,
                                                 const float* __restrict__ b_out,
                                                 float* __restrict__ out) {
  const int wave = threadIdx.x >> 5;   // 0..5 : output col tile
  const int lane = threadIdx.x & 31;
  const int hi   = lane >> 4;
  const int ln   = lane & 15;
  const long mt  = blockIdx.x;         // 0..8191 row tiles
  const int col  = wave * 16 + ln;
  const bool colok = col < RNN_O;

  const float* arow = hid + (mt * 16 + ln) * RNN_H + 2 * hi;
  const float* wrow = W_out + (long)(colok ? col : 0) * RNN_H + 2 * hi;
  const float bias = colok ? b_out[col] : 0.f;

  v8f c;
#pragma unroll
  for (int v = 0; v < 8; ++v) c[v] = bias;

#pragma unroll
  for (int ks = 0; ks < RNN_H / 4; ++ks) {                    // 64 k-steps
    v2f a = *(const v2f*)(arow + ks * 4);
    v2f b = {0.f, 0.f};
    if (colok) b = *(const v2f*)(wrow + ks * 4);
    c = wmma_f32_k4(a, b, c);
  }

  if (colok) {
    float* orow = out + (mt * 16 + 8 * hi) * RNN_O + col;
#pragma unroll
    for (int v = 0; v < 8; ++v) orow[(long)v * RNN_O] = c[v];
  }
}

// ---------------------------------------------------------------------------
extern "C" void kernel_launch(void* const* d_in, const int* in_sizes, int n_in,
                              void* d_out, int out_size, void* d_ws, size_t ws_size,
                              hipStream_t stream) {
  const float* x     = (const float*)d_in[0];
  const float* h0    = (const float*)d_in[1];
  const float* W_ih  = (const float*)d_in[2];
  const float* b_ih  = (const float*)d_in[3];
  const float* W_hh  = (const float*)d_in[4];
  const float* W_out = (const float*)d_in[5];
  const float* b_out = (const float*)d_in[6];

  float* out = (float*)d_out;                                  // [N,T,O]
  float* hid = out + (size_t)RNN_N * RNN_T * RNN_O;            // [N,T,H]

  const int row_tiles = (RNN_N * RNN_T) / 16;                  // 8192

  k_proj_u<<<dim3(row_tiles), dim3(512), 0, stream>>>(x, W_ih, b_ih, hid);
  k_scan<<<dim3(1), dim3(512), 0, stream>>>(h0, W_hh, hid);
  k_readout<<<dim3(row_tiles), dim3(192), 0, stream>>>(hid, W_out, b_out, out);
}